// MaterialHead_18674517803552
// MI455X (gfx1250) — compile-verified
//
#include <hip/hip_runtime.h>
#include <hip/hip_bf16.h>

typedef __attribute__((ext_vector_type(2))) float v2f;
typedef __attribute__((ext_vector_type(4))) float v4f;
typedef __attribute__((ext_vector_type(8))) float v8f;

#define FDIM 1024
#define HDIM 21
#define TASKID 3
#define KC 256                 // features per LDS chunk
#define NCHUNK (FDIM / KC)
#define WPB 8                  // waves per block
#define TPB (WPB * 32)
#define LDSW_FLOATS ((KC / 4) * 128)   // 8192 floats = 32 KB per chunk
#define WPACK_FLOATS (FDIM * 32)       // 32768 floats = 128 KB packed W1

__global__ void mh_zero(int* cnt) { if (threadIdx.x == 0) *cnt = 0; }

// Pre-pack W1 [1024,21] -> [k/4][n:32 (zero-padded)][k%4] so the per-chunk
// LDS fill in the MLP kernel is a plain contiguous copy.
__global__ __launch_bounds__(256) void mh_packW1(const float* __restrict__ W1,
                                                 float* __restrict__ Wp)
{
    int p = blockIdx.x * 256 + threadIdx.x;      // 0 .. FDIM*32-1
    int kgroup = p >> 7;                         // k/4
    int rem    = p & 127;
    int nn     = rem >> 2;                       // padded column 0..31
    int kk     = rem & 3;                        // k%4
    int kloc   = (kgroup << 2) | kk;
    Wp[p] = (nn < HDIM) ? W1[kloc * HDIM + nn] : 0.0f;
}

// Pass 1: default output = x2, compact active row indices (wave32 ballot).
__global__ __launch_bounds__(256) void mh_compact(
    const long long* __restrict__ task_ids, const float* __restrict__ x2,
    float* __restrict__ out, int* __restrict__ cnt, int* __restrict__ idx, int n)
{
    int i = blockIdx.x * blockDim.x + threadIdx.x;
    if (i >= n) return;
    out[i] = x2[i];
    bool active = (task_ids[i] == (long long)TASKID);
    unsigned ball = (unsigned)__ballot(active);   // wave32: low 32 bits
    unsigned lane = threadIdx.x & 31u;
    int nact = __popc(ball);
    if (nact == 0) return;
    int leader = __ffs(ball) - 1;
    int base = 0;
    if ((int)lane == leader) base = atomicAdd(cnt, nact);
    base = __shfl(base, leader, 32);
    if (active) {
        int pre = __popc(ball & ((1u << lane) - 1u));
        idx[base + pre] = i;
    }
}

// Pass 2: one wave per 16 compacted rows. fp32 WMMA 16x16x4:
//   A (16x4): lanes 0-15 hold M=lane, V0/V1 = K {0,1}; lanes 16-31 = K {2,3}
//   B (4x16): lanes 0-15 hold N=lane, V0/V1 = K {0,1}; lanes 16-31 = K {2,3}
//   C/D (16x16): VGPR v, lanes 0-15 -> (M=v, N=lane); lanes 16-31 -> (M=v+8)
__global__ __launch_bounds__(TPB) void mh_mlp(
    const float* __restrict__ x0, const float* __restrict__ Wp,
    const float* __restrict__ b1, const float* __restrict__ W2,
    const float* __restrict__ b2, const int* __restrict__ cnt,
    const int* __restrict__ idx, float* __restrict__ out, int n)
{
    __shared__ __align__(16) float ldsW[LDSW_FLOATS];  // 32 KB chunk of packed W1
    __shared__ float ldsY[WPB][16];

    int count = *cnt;
    int tile0 = blockIdx.x * WPB;
    if (tile0 * 16 >= count) return;         // uniform block early-out

    int tid   = threadIdx.x;
    int wave  = tid >> 5;
    int lane  = tid & 31;
    int mrow  = lane & 15;                   // M for A, N for B/C
    int klane = (lane >> 4) << 1;            // 0 for lanes 0-15, 2 for 16-31
    int tile  = tile0 + wave;

    // gathered per-lane row pointer (clamped so all waves keep running barriers)
    int slot  = tile * 16 + mrow;
    int cslot = slot < count ? slot : count - 1;
    int row   = idx[cslot];
    row = row < 0 ? 0 : (row >= n ? n - 1 : row);
    const float* arow = x0 + (size_t)row * FDIM;

    // accumulators init with b1 broadcast per column N
    float bias0 = b1[mrow];
    float bias1 = (mrow + 16 < HDIM) ? b1[mrow + 16] : 0.0f;
    v8f acc0, acc1;
#pragma unroll
    for (int v = 0; v < 8; ++v) { acc0[v] = bias0; acc1[v] = bias1; }

    for (int c = 0; c < NCHUNK; ++c) {
        int f0 = c * KC;
        __syncthreads();
        // contiguous 32 KB copy: stage 8 x b128 in regs, then 8 x ds_store_b128
        {
            const v4f* src = (const v4f*)(Wp + c * LDSW_FLOATS);
            v4f* dst = (v4f*)ldsW;
            v4f tmp[LDSW_FLOATS / 4 / TPB];                   // 8 per thread
#pragma unroll
            for (int j = 0; j < LDSW_FLOATS / 4 / TPB; ++j)
                tmp[j] = src[tid + j * TPB];
#pragma unroll
            for (int j = 0; j < LDSW_FLOATS / 4 / TPB; ++j)
                dst[tid + j * TPB] = tmp[j];
        }
        __syncthreads();
#pragma unroll 8
        for (int ks = 0; ks < KC; ks += 4) {
            v2f a = *(const v2f*)(arow + f0 + ks + klane);                 // global_load_b64
            const float* bb = &ldsW[(ks >> 2) * 128 + klane];
            v2f bt0 = *(const v2f*)(bb + mrow * 4);                        // ds_load_b64
            v2f bt1 = *(const v2f*)(bb + (mrow + 16) * 4);                 // ds_load_b64
            acc0 = __builtin_amdgcn_wmma_f32_16x16x4_f32(false, a, false, bt0,
                                                         (short)0, acc0, false, false);
            acc1 = __builtin_amdgcn_wmma_f32_16x16x4_f32(false, a, false, bt1,
                                                         (short)0, acc1, false, false);
        }
    }

    // layer 2: y = gelu(h) @ W2 + b2 ; column N lives in lane (N | half)
    float w2a = W2[mrow];
    float w2b = (mrow + 16 < HDIM) ? W2[mrow + 16] : 0.0f;
    float bb2 = b2[0];
#pragma unroll
    for (int v = 0; v < 8; ++v) {
        float h0 = acc0[v], h1 = acc1[v];
        float g0 = 0.5f * h0 * (1.0f + erff(h0 * 0.70710678118654752f));   // exact GELU
        float g1 = 0.5f * h1 * (1.0f + erff(h1 * 0.70710678118654752f));
        float p  = g0 * w2a + g1 * w2b;
        p += __shfl_xor(p, 8, 32);    // reduce N across the 16-lane half
        p += __shfl_xor(p, 4, 32);
        p += __shfl_xor(p, 2, 32);
        p += __shfl_xor(p, 1, 32);
        if (mrow == 0) {
            int M = v + ((lane >> 4) << 3);   // lanes 16-31 carry rows 8-15
            ldsY[wave][M] = p + bb2;
        }
    }
    __syncthreads();

    if (lane < 16) {
        int s = tile * 16 + lane;
        if (s < count) {
            int r = idx[s];
            if (r >= 0 && r < n) out[r] = ldsY[wave][lane];
        }
    }
}

extern "C" void kernel_launch(void* const* d_in, const int* in_sizes, int n_in,
                              void* d_out, int out_size, void* d_ws, size_t ws_size,
                              hipStream_t stream) {
    const float*     x0       = (const float*)d_in[0];
    const float*     x2       = (const float*)d_in[1];
    const float*     W1       = (const float*)d_in[2];
    const float*     b1       = (const float*)d_in[3];
    const float*     W2       = (const float*)d_in[4];
    const float*     b2       = (const float*)d_in[5];
    const long long* task_ids = (const long long*)d_in[6];
    float* out = (float*)d_out;
    int n = in_sizes[6];                 // N rows

    // d_ws layout: [cnt:4B pad to 256B][idx: n*4 B, pad to 256B][Wp: 128 KB]
    int*   cnt  = (int*)d_ws;
    int*   idxb = (int*)((char*)d_ws + 256);
    size_t idx_bytes = (((size_t)n * 4) + 255) & ~(size_t)255;
    float* Wp   = (float*)((char*)d_ws + 256 + idx_bytes);

    mh_zero<<<1, 1, 0, stream>>>(cnt);
    mh_packW1<<<WPACK_FLOATS / 256, 256, 0, stream>>>(W1, Wp);
    mh_compact<<<(n + 255) / 256, 256, 0, stream>>>(task_ids, x2, out, cnt, idxb, n);

    int tiles  = (n + 15) / 16;          // worst case: every row active
    int blocks = (tiles + WPB - 1) / WPB;
    mh_mlp<<<blocks, TPB, 0, stream>>>(x0, Wp, b1, W2, b2, cnt, idxb, out, n);
}